// GCNMasked_44487271252796
// MI455X (gfx1250) — compile-verified
//
#include <hip/hip_runtime.h>

// GCN (2-layer, masked edges) for MI455X / gfx1250, wave32.
// GEMMs use V_WMMA_F32_16X16X4_F32 (fp32-exact matrix path).
// Aggregation: float4 gathers from L2-resident h + global f32 atomics.

#define NNODES 50000
#define NEDGES 800000
#define INDIM  128
#define FDIM   64   // hidden == out == 64

typedef __attribute__((ext_vector_type(2))) float v2f;
typedef __attribute__((ext_vector_type(8))) float v8f;

// ---------------- degree / dinv ----------------

__global__ void deg_init_kernel(float* __restrict__ deg, int n) {
    int i = blockIdx.x * blockDim.x + threadIdx.x;
    if (i < n) deg[i] = 1.0f;                 // self-loop weight
}

__global__ void deg_accum_kernel(const long long* __restrict__ dst,
                                 const unsigned char* __restrict__ mask,
                                 float* __restrict__ deg, int E) {
    int e = blockIdx.x * blockDim.x + threadIdx.x;
    if (e < E && mask[e]) atomicAdd(&deg[(int)dst[e]], 1.0f);
}

__global__ void deg_to_dinv_kernel(float* __restrict__ deg, int n) {
    int i = blockIdx.x * blockDim.x + threadIdx.x;
    if (i < n) {
        float d = deg[i];
        deg[i] = (d > 0.0f) ? rsqrtf(d) : 0.0f;
    }
}

// ---------------- WMMA GEMM: C[M x 64] = A[M x K] * B[K x 64] ----------------
// One wave computes one 16x16 tile via V_WMMA_F32_16X16X4_F32.
// M = 50000 = 3125*16 exactly -> no ragged tiles, EXEC stays all-ones.

template <int K>
__global__ __launch_bounds__(128) void wmma_gemm_kernel(
    const float* __restrict__ A, const float* __restrict__ B,
    float* __restrict__ C, int M) {
    const int NOUT = FDIM;                       // 64
    int wave = threadIdx.x >> 5;
    int tile = blockIdx.x * 4 + wave;            // 4 waves per block
    int mt = tile >> 2;                          // NOUT/16 == 4 N-tiles
    int nt = tile & 3;
    if (mt * 16 >= M) return;                    // uniform per wave

    int lane = threadIdx.x & 31;
    int half = lane >> 4;                        // 0: K pair {0,1}; 1: {2,3}
    int lid  = lane & 15;

    const float* arow = A + (size_t)(mt * 16 + lid) * K;   // A row m = lid
    const float* bcol = B + nt * 16 + lid;                 // B col n = lid

    v8f acc = {};
#pragma unroll 8
    for (int k = 0; k < K; k += 4) {
        int ka = k + 2 * half;
        v2f a; a.x = arow[ka];                 a.y = arow[ka + 1];
        v2f b; b.x = bcol[(size_t)ka * NOUT];  b.y = bcol[(size_t)(ka + 1) * NOUT];
        acc = __builtin_amdgcn_wmma_f32_16x16x4_f32(
            /*neg_a=*/false, a, /*neg_b=*/false, b,
            /*c_mod=*/(short)0, acc, /*reuse_a=*/false, /*reuse_b=*/false);
    }
    // D layout: VGPR r -> row 8*half + r, col lid
    float* crow = C + (size_t)(mt * 16 + 8 * half) * NOUT + nt * 16 + lid;
#pragma unroll
    for (int r = 0; r < 8; ++r) crow[(size_t)r * NOUT] = acc[r];
}

// ---------------- self-loop init (optionally + bias) ----------------
// out[i,f] = dinv[i]^2 * h[i,f] (+ bias[f])

__global__ void self_init_kernel(const float* __restrict__ h,
                                 const float* __restrict__ dinv,
                                 const float* __restrict__ bias,
                                 float* __restrict__ out, int total) {
    int i = blockIdx.x * blockDim.x + threadIdx.x;
    if (i < total) {
        int node = i >> 6;                       // FDIM == 64
        int f    = i & 63;
        float di = dinv[node];
        float v  = di * di * h[i];
        if (bias) v += bias[f];
        out[i] = v;
    }
}

// ---------------- edge scatter: agg[dst] += dinv[s]*dinv[d] * h[src] --------
// 16 lanes per edge, each lane covers 4 features (B128 gather + 4 f32 atomics).
// h (12.8 MB) is L2-resident, so gathers and atomic RMWs resolve in L2.

__global__ __launch_bounds__(256) void edge_agg_kernel(
    const long long* __restrict__ src, const long long* __restrict__ dst,
    const unsigned char* __restrict__ mask, const float* __restrict__ dinv,
    const float* __restrict__ h, float* __restrict__ agg, int E) {
    int t  = blockIdx.x * 256 + threadIdx.x;
    int e  = t >> 4;
    int f4 = (t & 15) << 2;
    if (e >= E) return;
    if (!mask[e]) return;                        // weight-0 edge: no contribution
    int s = (int)src[e];
    int d = (int)dst[e];
    float c = dinv[s] * dinv[d];
    const float4 hv = *reinterpret_cast<const float4*>(h + (size_t)s * FDIM + f4);
    float* op = agg + (size_t)d * FDIM + f4;
    atomicAdd(op + 0, c * hv.x);
    atomicAdd(op + 1, c * hv.y);
    atomicAdd(op + 2, c * hv.z);
    atomicAdd(op + 3, c * hv.w);
}

// ---------------- bias + ReLU ----------------

__global__ void bias_relu_kernel(const float* __restrict__ agg,
                                 const float* __restrict__ bias,
                                 float* __restrict__ z, int total) {
    int i = blockIdx.x * blockDim.x + threadIdx.x;
    if (i < total) {
        float v = agg[i] + bias[i & 63];
        z[i] = v > 0.0f ? v : 0.0f;
    }
}

// ---------------- launch ----------------

extern "C" void kernel_launch(void* const* d_in, const int* in_sizes, int n_in,
                              void* d_out, int out_size, void* d_ws, size_t ws_size,
                              hipStream_t stream) {
    const float*         x    = (const float*)d_in[0];          // [N,128]
    const long long*     ei   = (const long long*)d_in[1];      // [2,E] int64
    const unsigned char* msk  = (const unsigned char*)d_in[2];  // [E] bool
    const float*         W1   = (const float*)d_in[3];          // [128,64]
    const float*         b1   = (const float*)d_in[4];          // [64]
    const float*         W2   = (const float*)d_in[5];          // [64,64]
    const float*         b2   = (const float*)d_in[6];          // [64]
    float*               out  = (float*)d_out;                  // [N,64]

    const int N = NNODES;
    const int E = in_sizes[2];                                  // 800000
    const long long* src = ei;
    const long long* dst = ei + E;

    // Workspace layout (floats): dinv[N] | h[N*64] | agg[N*64]  (~25.8 MB)
    float* dinv = (float*)d_ws;
    float* h    = dinv + N;                                     // 50000 floats -> 16B aligned
    float* agg  = h + (size_t)N * FDIM;

    const int T = 256;
    const int nf = N * FDIM;

    // degrees -> dinv (in place)
    deg_init_kernel<<<(N + T - 1) / T, T, 0, stream>>>(dinv, N);
    deg_accum_kernel<<<(E + T - 1) / T, T, 0, stream>>>(dst, msk, dinv, E);
    deg_to_dinv_kernel<<<(N + T - 1) / T, T, 0, stream>>>(dinv, N);

    // layer 1: h = x @ W1  (WMMA f32), then normalized aggregation
    wmma_gemm_kernel<INDIM><<<(N / 16) * 4 / 4, 128, 0, stream>>>(x, W1, h, N);
    self_init_kernel<<<(nf + T - 1) / T, T, 0, stream>>>(h, dinv, nullptr, agg, nf);
    edge_agg_kernel<<<((E * 16) + T - 1) / T, T, 0, stream>>>(src, dst, msk, dinv, h, agg, E);
    bias_relu_kernel<<<(nf + T - 1) / T, T, 0, stream>>>(agg, b1, h, nf);  // z1 -> h (reuse)

    // layer 2: h2 = z1 @ W2 -> agg (reuse), then aggregate straight into d_out
    wmma_gemm_kernel<FDIM><<<(N / 16) * 4 / 4, 128, 0, stream>>>(h, W2, agg, N);
    self_init_kernel<<<(nf + T - 1) / T, T, 0, stream>>>(agg, dinv, b2, out, nf);
    edge_agg_kernel<<<((E * 16) + T - 1) / T, T, 0, stream>>>(src, dst, msk, dinv, agg, out, E);
}